// QuantumNATEnhanced_65481071409711
// MI455X (gfx1250) — compile-verified
//
#include <hip/hip_runtime.h>
#include <math.h>

#define NQ     10
#define NL     3
#define NSTATE 1024          // 2^NQ
#define BATCH  16384
#define BN_EPS 1e-5f

#define NB     4             // sample-tiles (of 16) per wave in the GEMM kernel
#define NBS    (NB * 16)     // samples per wave = 64
#define ESTR   34            // padded LDS row stride (floats): 8B-aligned, bank-spread

typedef __attribute__((ext_vector_type(2))) float v2f;
typedef __attribute__((ext_vector_type(8))) float v8f;

#ifndef __has_builtin
#define __has_builtin(x) 0
#endif
#define HAVE_WMMA_F32X4 __has_builtin(__builtin_amdgcn_wmma_f32_16x16x4_f32)

// ---------------------------------------------------------------------------
// Kernel 1: build the fixed circuit unitary U (rotation layers + CNOT chains).
// One wave (32 lanes) per basis column; the 1024-complex state lives in
// 64 VGPRs: lane l holds amplitudes j = m*32 + l, m = 0..31.
// Qubit bit position p = 9 - q of the state index:
//   p >= 5 -> register-pair math (compile-time indices)
//   p <  5 -> cross-lane pairing via shfl_xor (wave32)
// ---------------------------------------------------------------------------

template <int P>
__device__ __forceinline__ void apply1q(float (&ar)[32], float (&ai)[32],
                                        float u00r, float u00i, float u01r, float u01i,
                                        float u10r, float u10i, float u11r, float u11i) {
  if constexpr (P >= 5) {
    constexpr int mm = 1 << (P - 5);
#pragma unroll
    for (int m = 0; m < 32; ++m) {
      if (!(m & mm)) {
        const float x0r = ar[m],      x0i = ai[m];
        const float x1r = ar[m | mm], x1i = ai[m | mm];
        ar[m]      = u00r * x0r - u00i * x0i + u01r * x1r - u01i * x1i;
        ai[m]      = u00r * x0i + u00i * x0r + u01r * x1i + u01i * x1r;
        ar[m | mm] = u10r * x0r - u10i * x0i + u11r * x1r - u11i * x1i;
        ai[m | mm] = u10r * x0i + u10i * x0r + u11r * x1i + u11i * x1r;
      }
    }
  } else {
    const int lane = (int)(threadIdx.x & 31u);
    const int bit  = (lane >> P) & 1;
    const float c0r = bit ? u10r : u00r, c0i = bit ? u10i : u00i;
    const float c1r = bit ? u11r : u01r, c1i = bit ? u11i : u01i;
#pragma unroll
    for (int m = 0; m < 32; ++m) {
      const float pr = __shfl_xor(ar[m], 1 << P, 32);
      const float pi = __shfl_xor(ai[m], 1 << P, 32);
      const float v0r = bit ? pr : ar[m], v0i = bit ? pi : ai[m];
      const float v1r = bit ? ar[m] : pr, v1i = bit ? ai[m] : pi;
      ar[m] = c0r * v0r - c0i * v0i + c1r * v1r - c1i * v1i;
      ai[m] = c0r * v0i + c0i * v0r + c1r * v1i + c1i * v1r;
    }
  }
}

// CNOT with control bit PC, target bit PT (state-index bit positions, PC = PT+1)
template <int PC, int PT>
__device__ __forceinline__ void cnot_gate(float (&ar)[32], float (&ai)[32]) {
  if constexpr (PC >= 5 && PT >= 5) {
    constexpr int cm = 1 << (PC - 5);
    constexpr int tm = 1 << (PT - 5);
#pragma unroll
    for (int m = 0; m < 32; ++m) {
      if ((m & cm) && !(m & tm)) {
        float t;
        t = ar[m]; ar[m] = ar[m | tm]; ar[m | tm] = t;
        t = ai[m]; ai[m] = ai[m | tm]; ai[m | tm] = t;
      }
    }
  } else if constexpr (PC >= 5) {  // PC==5 (m bit 0), PT==4 (lane bit 4)
#pragma unroll
    for (int m = 1; m < 32; m += 2) {
      ar[m] = __shfl_xor(ar[m], 16, 32);
      ai[m] = __shfl_xor(ai[m], 16, 32);
    }
  } else {  // both bits inside the lane index
    const int lane = (int)(threadIdx.x & 31u);
    const int src  = ((lane >> PC) & 1) ? (lane ^ (1 << PT)) : lane;
#pragma unroll
    for (int m = 0; m < 32; ++m) {
      ar[m] = __shfl(ar[m], src, 32);
      ai[m] = __shfl(ai[m], src, 32);
    }
  }
}

__global__ void __launch_bounds__(32) build_unitary(const float* __restrict__ params,
                                                    float* __restrict__ Ur,
                                                    float* __restrict__ Ui) {
  const int w    = (int)blockIdx.x;            // basis column index
  const int lane = (int)(threadIdx.x & 31u);

  float ar[32], ai[32];
#pragma unroll
  for (int m = 0; m < 32; ++m) {
    ar[m] = (m == (w >> 5) && lane == (w & 31)) ? 1.0f : 0.0f;
    ai[m] = 0.0f;
  }

#define ROT_Q(Q)                                                              \
  do {                                                                        \
    const float tz = params[((l * NQ + (Q)) * 3) + 0];                        \
    const float tx = params[((l * NQ + (Q)) * 3) + 1];                        \
    const float ty = params[((l * NQ + (Q)) * 3) + 2];                        \
    const float cz = cosf(0.5f * tz), sz = sinf(0.5f * tz);                   \
    apply1q<9 - (Q)>(ar, ai, cz, -sz, 0.f, 0.f, 0.f, 0.f, cz, sz);            \
    const float cx = cosf(0.5f * tx), sx = sinf(0.5f * tx);                   \
    apply1q<9 - (Q)>(ar, ai, cx, 0.f, 0.f, -sx, 0.f, -sx, cx, 0.f);           \
    const float cy = cosf(0.5f * ty), sy = sinf(0.5f * ty);                   \
    apply1q<9 - (Q)>(ar, ai, cy, 0.f, -sy, 0.f, sy, 0.f, cy, 0.f);            \
  } while (0)

  for (int l = 0; l < NL; ++l) {
    ROT_Q(0); ROT_Q(1); ROT_Q(2); ROT_Q(3); ROT_Q(4);
    ROT_Q(5); ROT_Q(6); ROT_Q(7); ROT_Q(8); ROT_Q(9);
    cnot_gate<9, 8>(ar, ai); cnot_gate<8, 7>(ar, ai); cnot_gate<7, 6>(ar, ai);
    cnot_gate<6, 5>(ar, ai); cnot_gate<5, 4>(ar, ai); cnot_gate<4, 3>(ar, ai);
    cnot_gate<3, 2>(ar, ai); cnot_gate<2, 1>(ar, ai); cnot_gate<1, 0>(ar, ai);
  }
#undef ROT_Q

  // Store U row-major over k: Ur[j*NSTATE + k] = <j|U|k>, column k == w.
#pragma unroll
  for (int m = 0; m < 32; ++m) {
    const int j = m * 32 + lane;
    Ur[(size_t)j * NSTATE + w] = ar[m];
    Ui[(size_t)j * NSTATE + w] = ai[m];
  }
}

// ---------------------------------------------------------------------------
// Kernel 2: fused encode -> (complex x real) GEMM via V_WMMA_F32_16X16X4_F32
//           -> |amp|^2 -> signed <Z_q> reductions.
//
// Transposed roles vs round 1: A = U tile (M = amplitude row j), B = encoding
// tile (N = sample). One wave handles NB=4 sample-tiles (64 samples), reusing
// each U fragment for 4 WMMAs per matrix -> 4x less L2 traffic on U
// (~17 FLOP per L2 byte instead of ~4).
//
// Encoding is stored factorized (Kronecker split): enc[b][k] =
// encHi[b][k>>5] * encLo[b][k&31]  -> 17 KB LDS instead of 256 KB.
//
// Fragment layouts per ISA 7.12.2 (32-bit, wave32):
//   A 16x4 : lanes 0-15 M=0..15, v0/v1 = K0/K1 ; lanes 16-31: v0/v1 = K2/K3
//   B 4x16 : lanes 0-15 N=0..15, v0/v1 = K0/K1 ; lanes 16-31: v0/v1 = K2/K3
//   D 16x16: reg r -> rows r (lanes 0-15) / r+8 (lanes 16-31), N = lane%16
//
// Because M is the amplitude index, the Z-sign for qubit q (bit p = 9-q of
// j = jt*16 + r + 8*half) is: p>=4 -> scalar (jt bits); p==3 -> half;
// p<=2 -> compile-time from r. So per tile we only need the row-sum of the
// 8 probabilities for q=0..6, and per-r compile-time signs for q=7..9.
// ---------------------------------------------------------------------------
__global__ void __launch_bounds__(32) gemm_expect(const float* __restrict__ x,
                                                  const float* __restrict__ Ur,
                                                  const float* __restrict__ Ui,
                                                  float* __restrict__ oc) {
  __shared__ float encHi[NBS][ESTR];  // [sample][k>>5],  5 high qubit bits
  __shared__ float encLo[NBS][ESTR];  // [sample][k&31],  5 low  qubit bits

  const int lane = (int)(threadIdx.x & 31u);
  const int n    = lane & 15;
  const int half = lane >> 4;
  const int b0   = (int)blockIdx.x * NBS;

  // --- Build factorized encoding: lane handles column h = lane (0..31) ---
  for (int b = 0; b < NBS; ++b) {
    float c[NQ], s[NQ];
#pragma unroll
    for (int q = 0; q < NQ; ++q) {
      const float t = 0.5f * x[(size_t)(b0 + b) * NQ + q];
      c[q] = cosf(t);
      s[q] = sinf(t);
    }
    float hi = 1.0f, lo = 1.0f;
#pragma unroll
    for (int q = 0; q < 5; ++q)   // qubits 0..4 -> bits 9..5 of k
      hi *= ((lane >> (4 - q)) & 1) ? s[q] : c[q];
#pragma unroll
    for (int q = 5; q < NQ; ++q)  // qubits 5..9 -> bits 4..0 of k
      lo *= ((lane >> (9 - q)) & 1) ? s[q] : c[q];
    encHi[b][lane] = hi;
    encLo[b][lane] = lo;
  }
  __syncthreads();

  float ex[NB][NQ];
#pragma unroll
  for (int t = 0; t < NB; ++t)
#pragma unroll
    for (int q = 0; q < NQ; ++q) ex[t][q] = 0.0f;

  const float halfsign = half ? -1.0f : 1.0f;  // q=6 sign (bit 3 of the row)

  for (int jt = 0; jt < NSTATE / 16; ++jt) {
    // A fragment rows: j = jt*16 + n, K slice selected by half
    const float* arow_re = Ur + (size_t)(jt * 16 + n) * NSTATE + 2 * half;
    const float* arow_im = Ui + (size_t)(jt * 16 + n) * NSTATE + 2 * half;

    v8f accr[NB], acci[NB];
#pragma unroll
    for (int t = 0; t < NB; ++t) {
      accr[t] = (v8f){0.f, 0.f, 0.f, 0.f, 0.f, 0.f, 0.f, 0.f};
      acci[t] = (v8f){0.f, 0.f, 0.f, 0.f, 0.f, 0.f, 0.f, 0.f};
    }

#pragma unroll 8
    for (int kt = 0; kt < NSTATE / 4; ++kt) {
      const int k0 = kt * 4;
      const v2f a_re = *(const v2f*)(arow_re + k0);     // global b64
      const v2f a_im = *(const v2f*)(arow_im + k0);     // global b64
      const int hiIdx = k0 >> 5;                        // constant across 8 kt
      const int loIdx = (k0 & 31) + 2 * half;
#pragma unroll
      for (int t = 0; t < NB; ++t) {
        const int smp = t * 16 + n;
        const float hf = encHi[smp][hiIdx];             // ds b32 (CSE'd)
        const v2f lf   = *(const v2f*)&encLo[smp][loIdx]; // ds b64, 8B aligned
        const v2f b    = lf * hf;
#if HAVE_WMMA_F32X4
        accr[t] = __builtin_amdgcn_wmma_f32_16x16x4_f32(false, a_re, false, b,
                                                        (short)0, accr[t], false, false);
        acci[t] = __builtin_amdgcn_wmma_f32_16x16x4_f32(false, a_im, false, b,
                                                        (short)0, acci[t], false, false);
#else
        accr[t][0] += a_re.x * b.x + a_re.y * b.y;  // placeholder (no wmma)
        acci[t][0] += a_im.x * b.x + a_im.y * b.y;
#endif
      }
    }

    // probs + signed Z accumulation
#pragma unroll
    for (int t = 0; t < NB; ++t) {
      float p[8];
#pragma unroll
      for (int r = 0; r < 8; ++r)
        p[r] = accr[t][r] * accr[t][r] + acci[t][r] * acci[t][r];

      float rowsum = 0.0f;
#pragma unroll
      for (int r = 0; r < 8; ++r) rowsum += p[r];

      // q = 0..5: sign from jt bit (5-q)  (scalar, uniform)
#pragma unroll
      for (int q = 0; q < 6; ++q)
        ex[t][q] += ((jt >> (5 - q)) & 1) ? -rowsum : rowsum;
      // q = 6: sign from the half (bit 3 of the row)
      ex[t][6] += halfsign * rowsum;
      // q = 7..9: sign from bit (9-q) of r (compile-time)
#pragma unroll
      for (int r = 0; r < 8; ++r) {
        ex[t][7] += ((r >> 2) & 1) ? -p[r] : p[r];
        ex[t][8] += ((r >> 1) & 1) ? -p[r] : p[r];
        ex[t][9] += ((r >> 0) & 1) ? -p[r] : p[r];
      }
    }
  }

  // Combine the two row-halves (lanes n and n+16 hold the same sample)
#pragma unroll
  for (int t = 0; t < NB; ++t)
#pragma unroll
    for (int q = 0; q < NQ; ++q)
      ex[t][q] += __shfl_xor(ex[t][q], 16, 32);

  if (half == 0) {
#pragma unroll
    for (int t = 0; t < NB; ++t) {
      const int b = b0 + t * 16 + n;
#pragma unroll
      for (int q = 0; q < NQ; ++q) oc[(size_t)b * NQ + q] = ex[t][q];
    }
  }
}

// ---------------------------------------------------------------------------
// Kernel 3: BatchNorm1d (training mode, biased variance) over the batch dim.
// ---------------------------------------------------------------------------
__global__ void __launch_bounds__(256) batchnorm_kernel(const float* __restrict__ oc,
                                                        const float* __restrict__ gamma,
                                                        const float* __restrict__ beta,
                                                        float* __restrict__ out) {
  __shared__ float s1[256];
  __shared__ float s2[256];
  const int q   = (int)blockIdx.x;
  const int tid = (int)threadIdx.x;

  float sum = 0.0f, sq = 0.0f;
  for (int b = tid; b < BATCH; b += 256) {
    const float v = oc[(size_t)b * NQ + q];
    sum += v;
    sq  += v * v;
  }
  s1[tid] = sum;
  s2[tid] = sq;
  __syncthreads();
  for (int off = 128; off > 0; off >>= 1) {
    if (tid < off) {
      s1[tid] += s1[tid + off];
      s2[tid] += s2[tid + off];
    }
    __syncthreads();
  }
  const float mean  = s1[0] * (1.0f / BATCH);
  const float var   = s2[0] * (1.0f / BATCH) - mean * mean;
  const float scale = gamma[q] * rsqrtf(var + BN_EPS);
  const float bias  = beta[q];
  for (int b = tid; b < BATCH; b += 256) {
    out[(size_t)b * NQ + q] = scale * (oc[(size_t)b * NQ + q] - mean) + bias;
  }
}

// ---------------------------------------------------------------------------
extern "C" void kernel_launch(void* const* d_in, const int* in_sizes, int n_in,
                              void* d_out, int out_size, void* d_ws, size_t ws_size,
                              hipStream_t stream) {
  (void)in_sizes; (void)n_in; (void)out_size; (void)ws_size;
  const float* x      = (const float*)d_in[0];
  const float* params = (const float*)d_in[1];
  const float* gamma  = (const float*)d_in[2];
  const float* beta   = (const float*)d_in[3];
  float* out = (float*)d_out;

  // Workspace: Ur (4MB) | Ui (4MB) | oc (640KB)   (needs ~8.7 MB)
  float* Ur = (float*)d_ws;
  float* Ui = Ur + (size_t)NSTATE * NSTATE;
  float* oc = Ui + (size_t)NSTATE * NSTATE;

  build_unitary<<<NSTATE, 32, 0, stream>>>(params, Ur, Ui);
  gemm_expect<<<BATCH / NBS, 32, 0, stream>>>(x, Ur, Ui, oc);
  batchnorm_kernel<<<NQ, 256, 0, stream>>>(oc, gamma, beta, out);
}